// SelfAttention_2001454760717
// MI455X (gfx1250) — compile-verified
//
#include <hip/hip_runtime.h>
#include <hip/hip_bf16.h>

typedef __attribute__((ext_vector_type(2))) float v2f;
typedef __attribute__((ext_vector_type(8))) float v8f;

#define B_  4
#define T_  64
#define N_  325
#define F_  3
#define HD_ 16   // d per head
#define NH_ 4    // heads
#define D_  64   // NH_*HD_
#define NPAD_ 328  // N_ rounded up to multiple of 4 for K=4 WMMA chunks

// ---------------------------------------------------------------- kernel 1
// S = adj + 2*adj@adj ; also zero the layernorm stats (replay-safe).
__global__ __launch_bounds__(256) void k_supports(const float* __restrict__ adj,
                                                  float* __restrict__ S,
                                                  float* __restrict__ stats) {
  int idx = blockIdx.x * 256 + threadIdx.x;
  if (blockIdx.x == 0 && threadIdx.x < 8) stats[threadIdx.x] = 0.0f;
  if (idx >= N_ * N_) return;
  int i = idx / N_, j = idx % N_;
  const float* ai = adj + (size_t)i * N_;
  float acc = 0.0f;
  for (int k = 0; k < N_; ++k) acc = fmaf(ai[k], adj[(size_t)k * N_ + j], acc);
  S[idx] = adj[idx] + 2.0f * acc;
}

// ---------------------------------------------------------------- kernel 2
// X[b,t,n,o] = relu(sum_f inp[b,t,n,f]*w_lt[o,f] + b_lt[o])   (also rescon)
__global__ __launch_bounds__(256) void k_lintrans(const float* __restrict__ inp,
                                                  const float* __restrict__ w,
                                                  const float* __restrict__ bias,
                                                  float* __restrict__ X) {
  int idx = blockIdx.x * 256 + threadIdx.x;   // (b*T+t)*N+n
  if (idx >= B_ * T_ * N_) return;
  float f0 = inp[(size_t)idx * 3 + 0];
  float f1 = inp[(size_t)idx * 3 + 1];
  float f2 = inp[(size_t)idx * 3 + 2];
  float* out = X + (size_t)idx * HD_;
#pragma unroll
  for (int o = 0; o < HD_; ++o) {
    float v = bias[o] + f0 * w[o * 3 + 0] + f1 * w[o * 3 + 1] + f2 * w[o * 3 + 2];
    out[o] = fmaxf(v, 0.0f);
  }
}

// ---------------------------------------------------------------- kernel 3
// Per (b,t): lfs = S @ X[b,t] (325x16) fully on the WMMA pipe (82 K=4
// chunks, X zero-padded to 328 rows so garbage A columns hit zero B rows),
// then Q/K/V = relu(lfs@w + b) as 16x16x4 WMMA chains, scattered to
// head-major layout (h,B,N,T,16). 4 waves/block, one row-tile per wave
// per iteration; padded uniform loop keeps barriers convergent.
__global__ __launch_bounds__(128) void k_gcn_qkv(
    const float* __restrict__ S, const float* __restrict__ X,
    const float* __restrict__ wq, const float* __restrict__ bq,
    const float* __restrict__ wk, const float* __restrict__ bk,
    const float* __restrict__ wv, const float* __restrict__ bv,
    float* __restrict__ Qh, float* __restrict__ Kh, float* __restrict__ Vh) {
  __shared__ float Xl[NPAD_ * HD_];   // 5248 floats, rows 325..327 zeroed
  __shared__ float Wl[3 * 16 * D_];   // staged wq,wk,wv (3072 floats)
  __shared__ float Lf[4][16 * 20];    // per-wave lfs tile (C->A relayout)

  const int bt = blockIdx.x;          // b*T + t
  const int b = bt / T_, t = bt % T_;
  const int tid = threadIdx.x;
  const int wave = tid >> 5;
  const int lane = tid & 31;
  const int half = lane >> 4;
  const int lc = lane & 15;
  const int koff = half * 2;

  // stage X[b,t] (then zero tail rows; disjoint regions, one barrier)
  {
    const float4* Xg = (const float4*)(X + (size_t)bt * (N_ * HD_));
    float4* Xl4 = (float4*)Xl;
    for (int i = tid; i < (N_ * HD_) / 4; i += 128) Xl4[i] = Xg[i];
    for (int i = N_ * HD_ + tid; i < NPAD_ * HD_; i += 128) Xl[i] = 0.0f;
    const float4* q4 = (const float4*)wq;
    const float4* k4 = (const float4*)wk;
    const float4* v4 = (const float4*)wv;
    float4* Wl4 = (float4*)Wl;
    for (int i = tid; i < 256; i += 128) {
      Wl4[i] = q4[i];
      Wl4[256 + i] = k4[i];
      Wl4[512 + i] = v4[i];
    }
  }
  __syncthreads();

  const v8f vzero = {0.f, 0.f, 0.f, 0.f, 0.f, 0.f, 0.f, 0.f};
  const float* biasp[3] = {bq, bk, bv};
  float* dstp[3] = {Qh, Kh, Vh};

  for (int it = 0; it < 6; ++it) {
    const int tile = wave * 6 + it;
    const int i0 = tile * 16;
    const bool active = (tile < 21);

    v8f acc = vzero;
    if (active) {
      // A rows come straight from L2-resident S (each lane owns one row)
      const float* rowA = S + (size_t)(i0 + lc) * N_ + koff;
      __builtin_prefetch(rowA + (size_t)16 * N_, 0, 1);  // next tile's rows
      for (int kc = 0; kc < 82; ++kc) {
        const int k0 = kc * 4;
        v2f aa;
        aa.x = rowA[k0];
        aa.y = rowA[k0 + 1];
        v2f bb;  // B = X[k][dd], dd striped across lanes
        bb.x = Xl[(k0 + koff) * HD_ + lc];
        bb.y = Xl[(k0 + koff + 1) * HD_ + lc];
        acc = __builtin_amdgcn_wmma_f32_16x16x4_f32(
            false, aa, false, bb, (short)0, acc, false, false);
      }
    }
    __syncthreads();
    if (active) {  // C-layout -> row-major LDS
#pragma unroll
      for (int v = 0; v < 8; ++v) Lf[wave][(v + half * 8) * 20 + lc] = acc[v];
    }
    __syncthreads();
    if (active) {
      // reload lfs tile in A-layout
      v2f aL[4];
#pragma unroll
      for (int kc = 0; kc < 4; ++kc) {
        aL[kc].x = Lf[wave][lc * 20 + kc * 4 + koff];
        aL[kc].y = Lf[wave][lc * 20 + kc * 4 + koff + 1];
      }
#pragma unroll
      for (int m = 0; m < 3; ++m) {
        const float* Wm = Wl + m * (16 * D_);
        const float* bias = biasp[m];
        float* dst = dstp[m];
#pragma unroll
        for (int ct = 0; ct < NH_; ++ct) {   // head ct <-> cols ct*16..+15
          const float bval = bias[ct * 16 + lc];
          v8f o;
#pragma unroll
          for (int v = 0; v < 8; ++v) o[v] = bval;
#pragma unroll
          for (int kc = 0; kc < 4; ++kc) {
            v2f bb;
            bb.x = Wm[(kc * 4 + koff) * D_ + ct * 16 + lc];
            bb.y = Wm[(kc * 4 + koff + 1) * D_ + ct * 16 + lc];
            o = __builtin_amdgcn_wmma_f32_16x16x4_f32(
                false, aL[kc], false, bb, (short)0, o, false, false);
          }
#pragma unroll
          for (int v = 0; v < 8; ++v) {
            const int i = i0 + v + half * 8;  // node index
            if (i < N_) {
              const size_t oidx =
                  (((size_t)ct * B_ + b) * N_ + i) * (T_ * HD_) + t * HD_ + lc;
              dst[oidx] = fmaxf(o[v], 0.0f);
            }
          }
        }
      }
    }
  }
}

// ---------------------------------------------------------------- kernel 4
// One wave per (hh,b,n). scores=QK^T/4 via v_wmma_f32_16x16x4_f32, full-row
// softmax, post-softmax tril mask to 1e-9, attn@V via WMMA, relu, store to
// res laid out (B,T,N,D) with channel = hh*16+dd.
__global__ __launch_bounds__(32) void k_attention(const float* __restrict__ Qh,
                                                  const float* __restrict__ Kh,
                                                  const float* __restrict__ Vh,
                                                  float* __restrict__ res) {
  __shared__ float Kl[T_ * HD_];     // 1024 floats
  __shared__ float Vl[T_ * HD_];     // 1024 floats
  __shared__ float Al[16 * 68];      // attn row-tile staging (padded)

  const int blk = blockIdx.x;        // ((hh*B+b)*N+n)
  const int n = blk % N_;
  const int hb = blk / N_;
  const int b = hb % B_;
  const int hh = hb / B_;
  const size_t base = (size_t)blk * (T_ * HD_);

  const int lane = threadIdx.x;
  const int half = lane >> 4;        // 0: lanes 0-15, 1: lanes 16-31
  const int lc = lane & 15;
  const int koff = half * 2;         // A/B K-offset per half-wave (ISA layout)

  { // stage K and V tiles into LDS (1024 floats each)
    const float4* Kg = (const float4*)(Kh + base);
    const float4* Vg = (const float4*)(Vh + base);
    float4* Kl4 = (float4*)Kl;
    float4* Vl4 = (float4*)Vl;
    for (int i = lane; i < 256; i += 32) { Kl4[i] = Kg[i]; Vl4[i] = Vg[i]; }
  }
  __syncthreads();

  const float* Qbase = Qh + base;
  const v8f vzero = {0.f, 0.f, 0.f, 0.f, 0.f, 0.f, 0.f, 0.f};
  const float scale = 0.25f;         // 1/sqrt(16)

  for (int rt = 0; rt < 4; ++rt) {
    const int t0 = rt * 16;
    // A operand: Q rows t0..t0+15, K-chunks of 4 (full d=16 -> 4 chunks)
    v2f aq[4];
    const float* qrow = Qbase + (size_t)(t0 + lc) * HD_;
#pragma unroll
    for (int kc = 0; kc < 4; ++kc) {
      aq[kc].x = qrow[kc * 4 + koff];
      aq[kc].y = qrow[kc * 4 + koff + 1];
    }
    // scores: 4 column tiles of 16, each K=16 -> 4 WMMA chained
    v8f acc[4];
#pragma unroll
    for (int ct = 0; ct < 4; ++ct) {
      acc[ct] = vzero;
#pragma unroll
      for (int kc = 0; kc < 4; ++kc) {
        v2f bb;                      // B = K^T : B[k][col] = K[s0+col][k]
        const float* kr = Kl + (size_t)(ct * 16 + lc) * HD_ + kc * 4 + koff;
        bb.x = kr[0];
        bb.y = kr[1];
        acc[ct] = __builtin_amdgcn_wmma_f32_16x16x4_f32(
            false, aq[kc], false, bb, (short)0, acc[ct], false, false);
      }
    }
    // ---- softmax over all 64 columns (rows live in a 16-lane half-wave) ----
    float rmax[8], rsum[8];
#pragma unroll
    for (int v = 0; v < 8; ++v) {
      float m = fmaxf(fmaxf(acc[0][v], acc[1][v]), fmaxf(acc[2][v], acc[3][v]));
#pragma unroll
      for (int off = 1; off < 16; off <<= 1) m = fmaxf(m, __shfl_xor(m, off, 32));
      rmax[v] = m * scale;
    }
#pragma unroll
    for (int v = 0; v < 8; ++v) {
      float sum = 0.0f;
#pragma unroll
      for (int ct = 0; ct < 4; ++ct) {
        const float e = __expf(acc[ct][v] * scale - rmax[v]);
        acc[ct][v] = e;
        sum += e;
      }
#pragma unroll
      for (int off = 1; off < 16; off <<= 1) sum += __shfl_xor(sum, off, 32);
      rsum[v] = sum;
    }
    // ---- post-softmax causal mask, stage attn row-major in LDS ----
    __syncthreads();                 // protect Al reuse across rt iterations
#pragma unroll
    for (int ct = 0; ct < 4; ++ct) {
      const int s = ct * 16 + lc;
#pragma unroll
      for (int v = 0; v < 8; ++v) {
        const int rr = v + half * 8;         // local row
        float a = acc[ct][v] / rsum[v];
        a = (s <= t0 + rr) ? a : 1e-9f;
        Al[rr * 68 + s] = a;
      }
    }
    __syncthreads();
    // ---- out = attn @ V : 16x16 output, K=64 -> 16 WMMA chained ----
    v8f outc = vzero;
#pragma unroll
    for (int kc = 0; kc < 16; ++kc) {
      const int k0 = kc * 4 + koff;
      v2f aa;                        // A: rows = lc for both halves
      aa.x = Al[lc * 68 + k0];
      aa.y = Al[lc * 68 + k0 + 1];
      v2f bb;                        // B = V[k][dd]
      bb.x = Vl[k0 * HD_ + lc];
      bb.y = Vl[(k0 + 1) * HD_ + lc];
      outc = __builtin_amdgcn_wmma_f32_16x16x4_f32(
          false, aa, false, bb, (short)0, outc, false, false);
    }
    // relu + scatter into (B,T,N,D) with channel hh*16+dd
#pragma unroll
    for (int v = 0; v < 8; ++v) {
      const int t = t0 + v + half * 8;
      const size_t oidx = (((size_t)b * T_ + t) * N_ + n) * D_ + hh * 16 + lc;
      res[oidx] = fmaxf(outc[v], 0.0f);
    }
  }
}

// ---------------------------------------------------------------- kernel 5
// y = relu(w_ls@res + b_ls) + X ; accumulate per-batch sum / sumsq.
__global__ __launch_bounds__(256) void k_proj_res(const float* __restrict__ res,
                                                  const float* __restrict__ w_ls,
                                                  const float* __restrict__ b_ls,
                                                  const float* __restrict__ X,
                                                  float* __restrict__ Y,
                                                  float* __restrict__ stats) {
  __shared__ float red0[256];
  __shared__ float red1[256];
  const int b = blockIdx.y;
  const int flat = blockIdx.x * 256 + threadIdx.x;  // t*N+n
  float s0 = 0.0f, s1 = 0.0f;
  if (flat < T_ * N_) {
    const size_t idx = (size_t)b * T_ * N_ + flat;
    float rloc[D_];
    const float4* rv = (const float4*)(res + idx * D_);
#pragma unroll
    for (int q = 0; q < D_ / 4; ++q) ((float4*)rloc)[q] = rv[q];
    const float* xv = X + idx * HD_;
    float* yv = Y + idx * HD_;
#pragma unroll
    for (int o = 0; o < HD_; ++o) {
      float a = b_ls[o];
      const float* wrow = w_ls + o * D_;
#pragma unroll
      for (int c = 0; c < D_; ++c) a = fmaf(wrow[c], rloc[c], a);
      a = fmaxf(a, 0.0f) + xv[o];
      yv[o] = a;
      s0 += a;
      s1 += a * a;
    }
  }
  red0[threadIdx.x] = s0;
  red1[threadIdx.x] = s1;
  __syncthreads();
  for (int s = 128; s > 0; s >>= 1) {
    if (threadIdx.x < s) {
      red0[threadIdx.x] += red0[threadIdx.x + s];
      red1[threadIdx.x] += red1[threadIdx.x + s];
    }
    __syncthreads();
  }
  if (threadIdx.x == 0) {
    atomicAdd(&stats[b * 2 + 0], red0[0]);
    atomicAdd(&stats[b * 2 + 1], red1[0]);
  }
}

// ---------------------------------------------------------------- kernel 6
// Per-batch layernorm + gamma/beta, output (B,T,N,16).
__global__ __launch_bounds__(256) void k_norm(const float* __restrict__ Y,
                                              const float* __restrict__ stats,
                                              const float* __restrict__ gamma,
                                              const float* __restrict__ beta,
                                              float* __restrict__ out) {
  const int idx = blockIdx.x * 256 + threadIdx.x;
  if (idx >= B_ * T_ * N_ * HD_) return;
  const int o = idx & 15;
  const int bn = idx >> 4;             // (b*T+t)*N+n
  const int n = bn % N_;
  const int bt = bn / N_;
  const int t = bt % T_;
  const int b = bt / T_;
  const float M = (float)(HD_ * T_ * N_);
  const float mu = stats[b * 2 + 0] / M;
  const float var = stats[b * 2 + 1] / M - mu * mu;
  const float inv = rsqrtf(var + 1e-5f);
  const size_t gi = ((size_t)o * T_ + t) * N_ + n;
  out[idx] = (Y[idx] - mu) * inv * gamma[gi] + beta[gi];
}

// ---------------------------------------------------------------- launcher
extern "C" void kernel_launch(void* const* d_in, const int* in_sizes, int n_in,
                              void* d_out, int out_size, void* d_ws, size_t ws_size,
                              hipStream_t stream) {
  const float* inputs = (const float*)d_in[0];
  const float* adj    = (const float*)d_in[1];
  const float* w_lt   = (const float*)d_in[2];
  const float* b_lt   = (const float*)d_in[3];
  const float* wq     = (const float*)d_in[4];
  const float* bq     = (const float*)d_in[5];
  const float* wk     = (const float*)d_in[6];
  const float* bk     = (const float*)d_in[7];
  const float* wv     = (const float*)d_in[8];
  const float* bv     = (const float*)d_in[9];
  const float* w_ls   = (const float*)d_in[10];
  const float* b_ls   = (const float*)d_in[11];
  const float* gamma  = (const float*)d_in[12];
  const float* beta   = (const float*)d_in[13];
  float* out = (float*)d_out;

  char* ws = (char*)d_ws;
  auto alloc = [&](size_t nelem) {
    float* p = (float*)ws;
    ws += ((nelem * sizeof(float) + 255) / 256) * 256;
    return p;
  };
  const size_t BTN = (size_t)B_ * T_ * N_;
  float* S     = alloc((size_t)N_ * N_);
  float* X     = alloc(BTN * HD_);
  float* Qh    = alloc(BTN * D_);
  float* Kh    = alloc(BTN * D_);
  float* Vh    = alloc(BTN * D_);
  float* res   = alloc(BTN * D_);
  float* Y     = alloc(BTN * HD_);
  float* stats = alloc(8);

  k_supports<<<(N_ * N_ + 255) / 256, 256, 0, stream>>>(adj, S, stats);
  k_lintrans<<<(int)((BTN + 255) / 256), 256, 0, stream>>>(inputs, w_lt, b_lt, X);
  k_gcn_qkv<<<B_ * T_, 128, 0, stream>>>(S, X, wq, bq, wk, bk, wv, bv, Qh, Kh, Vh);
  k_attention<<<NH_ * B_ * N_, 32, 0, stream>>>(Qh, Kh, Vh, res);
  dim3 g5((T_ * N_ + 255) / 256, B_);
  k_proj_res<<<g5, 256, 0, stream>>>(res, w_ls, b_ls, X, Y, stats);
  k_norm<<<(int)((BTN * HD_ + 255) / 256), 256, 0, stream>>>(Y, stats, gamma, beta, out);
}